// SparseExpertCountingNetwork_1125281431619
// MI455X (gfx1250) — compile-verified
//
#include <hip/hip_runtime.h>
#include <hip/hip_bf16.h>

// ---------------------------------------------------------------------------
// SparseExpertCountingNetwork for MI455X (gfx1250).
// Memory-bound streaming kernel: one pass over x computes the router logits
// (via V_WMMA_F32_16X16X4_F32, N padded 4->16) and all four expert scalars.
// ---------------------------------------------------------------------------

#define TOKENS   32768          // B*S = 8*4096
#define DDIM     2048
#define NEXP     4
#define EPSF     1e-6f
#define CHUNK    128            // K elements staged per chunk
#define CHUNKP   132            // padded LDS row stride (16B aligned, bank-safe)
#define NCHUNK   (DDIM / CHUNK) // 16
#define TPB      16             // tokens per block
#define THREADS  256            // 8 wave32

typedef __attribute__((ext_vector_type(2))) float v2f;
typedef __attribute__((ext_vector_type(8))) float v8f;

__global__ __launch_bounds__(THREADS)
void moe_count_kernel(const float* __restrict__ x,
                      const float* __restrict__ gumbel,
                      const float* __restrict__ W,
                      const float* __restrict__ bias,
                      float* __restrict__ out)
{
    __shared__ float ldsX[2][TPB * CHUNKP];     // ~16.5 KB double-buffered A tile
    __shared__ float ldsW[DDIM * NEXP];         // 32 KB router weights
    __shared__ float ldsPart[8 * TPB * NEXP];   // 2 KB per-wave logit partials
    __shared__ float ldsScal[TPB * 4];          // per-token {sum, uniq, pat}

    const int tid  = threadIdx.x;
    const int wave = tid >> 5;
    const int lane = tid & 31;
    const int col  = lane & 15;                 // N (and A row M) index
    const int half = lane >> 4;                 // 0: K pair {0,1}; 1: {2,3}
    const int tokBase = blockIdx.x * TPB;

    // Stage W[2048][4] into LDS (coalesced float4).
    for (int i = tid; i < (DDIM * NEXP) / 4; i += THREADS)
        ((float4*)ldsW)[i] = ((const float4*)W)[i];
    __syncthreads();

    // Scalar-expert accumulators for this wave's two tokens.
    float sSum[2]  = {0.f, 0.f};
    float sUniq[2] = {0.f, 0.f};
    float sPat[2]  = {0.f, 0.f};
    float carry[2] = {0.f, 0.f};

    v8f acc = {};                               // WMMA C accumulator (f32)

    int buf = 0;
    for (int c = 0; c < NCHUNK; ++c) {
        // ---- stream chunk: wave handles tokens 2w and 2w+1 ---------------
        #pragma unroll
        for (int j = 0; j < 2; ++j) {
            const int trow  = wave * 2 + j;
            const int token = tokBase + trow;
            const int dg    = c * CHUNK + lane * 4;
            const float4 v  = *(const float4*)(x + (size_t)token * DDIM + dg);
            if (c + 1 < NCHUNK)  // pull next chunk toward the caches
                __builtin_prefetch(x + (size_t)token * DDIM + dg + CHUNK, 0, 0);

            sSum[j]  += (v.x + v.y) + (v.z + v.w);
            sUniq[j] += (v.x != 0.f ? 1.f : 0.f) + (v.y != 0.f ? 1.f : 0.f)
                      + (v.z != 0.f ? 1.f : 0.f) + (v.w != 0.f ? 1.f : 0.f);

            // pattern: sum over k>0 of [x_k != x_{k-1}] * (D - k)
            float prevw  = __shfl_up(v.w, 1);
            float pfirst = (lane == 0) ? carry[j] : prevw;
            if (dg > 0 && v.x != pfirst) sPat[j] += (float)(DDIM - dg);
            if (v.y != v.x)              sPat[j] += (float)(DDIM - (dg + 1));
            if (v.z != v.y)              sPat[j] += (float)(DDIM - (dg + 2));
            if (v.w != v.z)              sPat[j] += (float)(DDIM - (dg + 3));
            carry[j] = __shfl(v.w, 31);

            *(float4*)(&ldsX[buf][trow * CHUNKP + lane * 4]) = v;
        }
        __syncthreads();   // single barrier is safe: LDS tile is double-buffered

        // ---- WMMA router: this wave covers K sub-range [w*16, w*16+16) ----
        #pragma unroll
        for (int q = 0; q < 4; ++q) {
            const int klocal = wave * 16 + q * 4 + half * 2;   // within chunk
            const int kglob  = c * CHUNK + klocal;
            // A (16x4 f32): lane holds token row M=col, K = klocal, klocal+1
            const float2 av = *(const float2*)(&ldsX[buf][col * CHUNKP + klocal]);
            v2f a;  a[0] = av.x;  a[1] = av.y;
            // B (4x16 f32): lane holds column N=col, same two K rows; pad N>=4 with 0
            v2f bm;
            bm[0] = (col < NEXP) ? ldsW[kglob * NEXP + col]       : 0.f;
            bm[1] = (col < NEXP) ? ldsW[(kglob + 1) * NEXP + col] : 0.f;
            acc = __builtin_amdgcn_wmma_f32_16x16x4_f32(
                      false, a, false, bm, (short)0, acc, false, false);
        }
        buf ^= 1;
    }

    // ---- deterministic wave-level reduction of scalar experts ------------
    #pragma unroll
    for (int j = 0; j < 2; ++j) {
        float a0 = sSum[j], a1 = sUniq[j], a2 = sPat[j];
        #pragma unroll
        for (int off = 16; off > 0; off >>= 1) {
            a0 += __shfl_xor(a0, off);
            a1 += __shfl_xor(a1, off);
            a2 += __shfl_xor(a2, off);
        }
        if (lane == 0) {
            const int trow = wave * 2 + j;
            ldsScal[trow * 4 + 0] = a0;
            ldsScal[trow * 4 + 1] = a1;
            ldsScal[trow * 4 + 2] = a2;
        }
    }

    // ---- publish per-wave WMMA partials (fixed-order reduce => deterministic)
    if (col < NEXP) {
        #pragma unroll
        for (int r = 0; r < 8; ++r) {
            const int m = r + half * 8;               // token row in C
            ldsPart[(wave * TPB + m) * NEXP + col] = acc[r];
        }
    }
    __syncthreads();

    // ---- finalize: 16 threads, one token each -----------------------------
    if (tid < TPB) {
        const int token = tokBase + tid;
        const float s = ldsScal[tid * 4 + 0];
        const float u = ldsScal[tid * 4 + 1];
        const float p = ldsScal[tid * 4 + 2];
        float ev[4];
        ev[0] = s;                                    // HistogramExpert
        ev[1] = s / ((float)DDIM * (s + EPSF));       // FrequencyExpert (mean)
        ev[2] = u;                                    // UniquenessExpert
        ev[3] = p / (float)DDIM;                      // PatternCountExpert

        int best = 0;
        float bestv = -INFINITY;
        #pragma unroll
        for (int e = 0; e < NEXP; ++e) {
            float le = 0.f;
            #pragma unroll
            for (int w = 0; w < 8; ++w)               // fixed summation order
                le += ldsPart[(w * TPB + tid) * NEXP + e];
            le += bias[e] + gumbel[(size_t)token * NEXP + e];
            if (le > bestv) { bestv = le; best = e; } // first-max tie-break
        }
        out[token] = ev[best];
    }
}

extern "C" void kernel_launch(void* const* d_in, const int* in_sizes, int n_in,
                              void* d_out, int out_size, void* d_ws, size_t ws_size,
                              hipStream_t stream) {
    (void)in_sizes; (void)n_in; (void)d_ws; (void)ws_size; (void)out_size;
    const float* x      = (const float*)d_in[0];
    const float* gumbel = (const float*)d_in[1];
    const float* W      = (const float*)d_in[2];
    const float* b      = (const float*)d_in[3];
    float* out          = (float*)d_out;

    dim3 grid(TOKENS / TPB);   // 2048 blocks x 256 threads
    moe_count_kernel<<<grid, THREADS, 0, stream>>>(x, gumbel, W, b, out);
}